// GATEncoder_30288109372230
// MI455X (gfx1250) — compile-verified
//
#include <hip/hip_runtime.h>
#include <math.h>

typedef __attribute__((ext_vector_type(2))) float v2f;
typedef __attribute__((ext_vector_type(8))) float v8f;

#define GAT_H 4
#define GAT_F 128   // H*C
#define GAT_EMB 64

// ---------------------------------------------------------------- utilities
__global__ void fill_f(float* __restrict__ p, float v, long n) {
  long i = (long)blockIdx.x * blockDim.x + threadIdx.x;
  long stride = (long)gridDim.x * blockDim.x;
  for (; i < n; i += stride) p[i] = v;
}

__device__ inline void atomicMaxF(float* addr, float val) {
  int old = __float_as_int(*addr);
  while (val > __int_as_float(old)) {
    int assumed = old;
    old = atomicCAS((int*)addr, assumed, __float_as_int(val));
    if (old == assumed) break;
  }
}

// ------------------------------------------------- request_quantity stats
__global__ void reduce_stats(const float* __restrict__ rq, int n, float* __restrict__ stats) {
  __shared__ float ssum[256];
  __shared__ float ssq[256];
  int tid = blockIdx.x * blockDim.x + threadIdx.x;
  float v = 0.f, v2 = 0.f;
  for (int i = tid; i < n; i += gridDim.x * blockDim.x) {
    float x = rq[i]; v += x; v2 += x * x;
  }
  ssum[threadIdx.x] = v; ssq[threadIdx.x] = v2;
  __syncthreads();
  for (int s = 128; s > 0; s >>= 1) {
    if (threadIdx.x < s) {
      ssum[threadIdx.x] += ssum[threadIdx.x + s];
      ssq[threadIdx.x]  += ssq[threadIdx.x + s];
    }
    __syncthreads();
  }
  if (threadIdx.x == 0) {
    atomicAdd(&stats[0], ssum[0]);
    atomicAdd(&stats[1], ssq[0]);
  }
}

// ----------------------------------------- x = concat(emb[label], rq*W+b)
__global__ void build_x(const int* __restrict__ label, const float* __restrict__ rq,
                        const float* __restrict__ emb, const float* __restrict__ reqW,
                        const float* __restrict__ reqb, const float* __restrict__ stats,
                        float* __restrict__ x, int n) {
  long tid = (long)blockIdx.x * blockDim.x + threadIdx.x;
  long total = (long)n * GAT_F;
  if (tid >= total) return;
  int node = (int)(tid >> 7);
  int j = (int)(tid & 127);
  float out;
  if (j < GAT_EMB) {
    out = emb[(long)label[node] * GAT_EMB + j];
  } else {
    int jj = j - GAT_EMB;
    float mean = stats[0] / (float)n;
    float var = (stats[1] - (float)n * mean * mean) / (float)(n - 1);
    var = var > 0.f ? var : 0.f;
    float sd = sqrtf(var);
    float rqn = (rq[node] - mean) / (sd + 1e-6f);
    out = rqn * reqW[jj] + reqb[jj];
  }
  x[tid] = out;
}

// ------------------------------------------------- h = x @ W via f32 WMMA
// One block = 256 threads = 8 waves. Wave w computes the 16x16 tile at
// columns [16w,16w+16) of the block's 16-row strip; K=128 -> 32 chained
// V_WMMA_F32_16X16X4_F32 ops. W (64KB) stays resident in WGP$/L2, and the
// compiler software-pipelines the fragment loads across the WMMA chain.
__global__ __launch_bounds__(256) void gemm128_wmma(const float* __restrict__ X,
                                                    const float* __restrict__ W,
                                                    float* __restrict__ Hout, int n) {
  int wave = threadIdx.x >> 5;
  int lane = threadIdx.x & 31;
  int m = lane & 15;
  int grp = lane >> 4;          // 0 or 1
  int row = blockIdx.x * 16 + m;
  if (row >= n) row = n - 1;    // clamp loads; stores guarded below
  int col = wave * 16 + m;
  const float* xrow = X + (long)row * GAT_F;
  const int koff = grp * 2;
  v8f acc = {};
#pragma unroll
  for (int kt = 0; kt < 32; ++kt) {
    int k0 = kt * 4 + koff;
    v2f a; a.x = xrow[k0];                   a.y = xrow[k0 + 1];
    v2f b; b.x = W[(long)k0 * GAT_F + col];  b.y = W[(long)(k0 + 1) * GAT_F + col];
    acc = __builtin_amdgcn_wmma_f32_16x16x4_f32(false, a, false, b,
                                                (short)0, acc, false, false);
  }
  int rbase = blockIdx.x * 16 + grp * 8;
#pragma unroll
  for (int v = 0; v < 8; ++v) {
    int r = rbase + v;
    if (r < n) Hout[(long)r * GAT_F + col] = acc[v];
  }
}

// ------------------------------------- attention logits per (node, head)
__global__ void attn_logits(const float* __restrict__ Hm, const float* __restrict__ a_src,
                            const float* __restrict__ a_dst, float* __restrict__ als,
                            float* __restrict__ ald, int n) {
  int tid = blockIdx.x * blockDim.x + threadIdx.x;
  if (tid >= n * GAT_H) return;
  int node = tid >> 2;
  int hh = tid & 3;
  const float* hrow = Hm + (long)node * GAT_F + hh * 32;
  const float* as = a_src + hh * 32;
  const float* ad = a_dst + hh * 32;
  float s0 = 0.f, s1 = 0.f;
#pragma unroll
  for (int c = 0; c < 32; ++c) {
    float hv = hrow[c];
    s0 += hv * as[c];
    s1 += hv * ad[c];
  }
  als[tid] = s0;
  ald[tid] = s1;
}

// --------- pass 1: e = leaky_relu(als[src]+ald[dst]), seg-max (per edge)
__global__ void edge_logits_max(const int* __restrict__ src, const int* __restrict__ dst,
                                int E, int n, const float* __restrict__ als,
                                const float* __restrict__ ald, float* __restrict__ ebuf,
                                float* __restrict__ mbuf) {
  int e = blockIdx.x * blockDim.x + threadIdx.x;
  int Etot = E + n;
  if (e >= Etot) return;
  int s = (e < E) ? src[e] : (e - E);
  int d = (e < E) ? dst[e] : (e - E);
  const float4 a4 = *(const float4*)(als + (long)s * GAT_H);
  const float4 b4 = *(const float4*)(ald + (long)d * GAT_H);
  float4 v;
  v.x = a4.x + b4.x; v.x = (v.x > 0.f) ? v.x : 0.2f * v.x;
  v.y = a4.y + b4.y; v.y = (v.y > 0.f) ? v.y : 0.2f * v.y;
  v.z = a4.z + b4.z; v.z = (v.z > 0.f) ? v.z : 0.2f * v.z;
  v.w = a4.w + b4.w; v.w = (v.w > 0.f) ? v.w : 0.2f * v.w;
  *(float4*)(ebuf + (long)e * GAT_H) = v;
  float* mrow = mbuf + (long)d * GAT_H;
  atomicMaxF(mrow + 0, v.x);
  atomicMaxF(mrow + 1, v.y);
  atomicMaxF(mrow + 2, v.z);
  atomicMaxF(mrow + 3, v.w);
}

// ------------- pass 2: p = exp(e - m[dst]), seg-sum into sbuf (per edge)
__global__ void edge_softmax_sum(const int* __restrict__ src, const int* __restrict__ dst,
                                 int E, int n, float* __restrict__ ebuf,
                                 const float* __restrict__ mbuf, float* __restrict__ sbuf) {
  int e = blockIdx.x * blockDim.x + threadIdx.x;
  int Etot = E + n;
  if (e >= Etot) return;
  int d = (e < E) ? dst[e] : (e - E);
  float4 ev = *(const float4*)(ebuf + (long)e * GAT_H);
  const float4 mv = *(const float4*)(mbuf + (long)d * GAT_H);
  float4 p;
  p.x = expf(ev.x - mv.x);
  p.y = expf(ev.y - mv.y);
  p.z = expf(ev.z - mv.z);
  p.w = expf(ev.w - mv.w);
  *(float4*)(ebuf + (long)e * GAT_H) = p;
  float* srow = sbuf + (long)d * GAT_H;
  atomicAdd(srow + 0, p.x);
  atomicAdd(srow + 1, p.y);
  atomicAdd(srow + 2, p.z);
  atomicAdd(srow + 3, p.w);
}

// --------------- pass 3: out[dst] += h[src] * alpha  (float4 per thread)
__global__ void edge_aggregate(const int* __restrict__ src, const int* __restrict__ dst,
                               int E, int n, const float* __restrict__ Hm,
                               const float* __restrict__ ebuf, const float* __restrict__ sbuf,
                               float* __restrict__ out) {
  long tid = (long)blockIdx.x * blockDim.x + threadIdx.x;
  long Etot = (long)E + n;
  if (tid >= Etot * 32) return;
  int e = (int)(tid >> 5);
  int q = (int)(tid & 31);
  int ch0 = q * 4;            // 4 contiguous channels, all in the same head
  int hh = ch0 >> 5;
  int s = (e < E) ? src[e] : (e - E);
  int d = (e < E) ? dst[e] : (e - E);
  float alpha = ebuf[(long)e * GAT_H + hh] / (sbuf[(long)d * GAT_H + hh] + 1e-16f);
  const float4 hv = *(const float4*)(Hm + (long)s * GAT_F + ch0);
  float* orow = out + (long)d * GAT_F + ch0;
  atomicAdd(orow + 0, hv.x * alpha);
  atomicAdd(orow + 1, hv.y * alpha);
  atomicAdd(orow + 2, hv.z * alpha);
  atomicAdd(orow + 3, hv.w * alpha);
}

// ----------------------------------------- out = GELU_exact(acc + bias)
__global__ void bias_gelu(const float* __restrict__ acc, const float* __restrict__ b,
                          float* __restrict__ out, long total) {
  long i = (long)blockIdx.x * blockDim.x + threadIdx.x;
  if (i >= total) return;
  float v = acc[i] + b[(int)(i & 127)];
  out[i] = 0.5f * v * (1.0f + erff(v * 0.70710678118654752f));
}

// ---------------------------------------------------------------- driver
extern "C" void kernel_launch(void* const* d_in, const int* in_sizes, int n_in,
                              void* d_out, int out_size, void* d_ws, size_t ws_size,
                              hipStream_t stream) {
  (void)n_in; (void)out_size; (void)ws_size;
  const int*   label = (const int*)d_in[0];
  const int*   eidx  = (const int*)d_in[1];
  const float* rq    = (const float*)d_in[3];   // d_in[2] (weight) unused
  const float* emb   = (const float*)d_in[4];
  const float* reqW  = (const float*)d_in[5];
  const float* reqb  = (const float*)d_in[6];
  const float* Wl[3]  = {(const float*)d_in[7],  (const float*)d_in[11], (const float*)d_in[15]};
  const float* asr[3] = {(const float*)d_in[8],  (const float*)d_in[12], (const float*)d_in[16]};
  const float* adt[3] = {(const float*)d_in[9],  (const float*)d_in[13], (const float*)d_in[17]};
  const float* bl[3]  = {(const float*)d_in[10], (const float*)d_in[14], (const float*)d_in[18]};

  const int N = in_sizes[0];
  const int E = in_sizes[1] / 2;
  const int Etot = E + N;
  const int* src = eidx;
  const int* dst = eidx + E;

  // workspace layout
  char* ws = (char*)d_ws;
  size_t off = 0;
  auto alloc = [&](size_t bytes) -> float* {
    float* p = (float*)(ws + off);
    off = (off + bytes + 255) & ~(size_t)255;
    return p;
  };
  const size_t nodeF = (size_t)N * GAT_F * sizeof(float);
  const size_t nodeH = (size_t)N * GAT_H * sizeof(float);
  float* stats = alloc(2 * sizeof(float));
  float* xA    = alloc(nodeF);
  float* xB    = alloc(nodeF);
  float* hb    = alloc(nodeF);
  float* als   = alloc(nodeH);
  float* ald   = alloc(nodeH);
  float* mb    = alloc(nodeH);
  float* sb    = alloc(nodeH);
  float* eb    = alloc((size_t)Etot * GAT_H * sizeof(float));

  const int T = 256;
  const long nodeElems = (long)N * GAT_F;
  const int nhElems = N * GAT_H;
  const long aggElems = (long)Etot * 32;

  // node features
  fill_f<<<1, T, 0, stream>>>(stats, 0.f, 2);
  reduce_stats<<<256, T, 0, stream>>>(rq, N, stats);
  build_x<<<(int)((nodeElems + T - 1) / T), T, 0, stream>>>(label, rq, emb, reqW, reqb,
                                                            stats, xA, N);

  float* xin = xA;
  for (int layer = 0; layer < 3; ++layer) {
    float* obuf = (layer == 2) ? (float*)d_out : ((layer == 0) ? xB : xA);
    float* xnext = obuf;  // gelu in place; becomes next layer's input

    gemm128_wmma<<<(N + 15) / 16, 256, 0, stream>>>(xin, Wl[layer], hb, N);
    attn_logits<<<(nhElems + T - 1) / T, T, 0, stream>>>(hb, asr[layer], adt[layer],
                                                         als, ald, N);
    fill_f<<<(nhElems + T - 1) / T, T, 0, stream>>>(mb, -3.0e38f, nhElems);
    fill_f<<<(nhElems + T - 1) / T, T, 0, stream>>>(sb, 0.f, nhElems);
    fill_f<<<(int)((nodeElems + T - 1) / T), T, 0, stream>>>(obuf, 0.f, nodeElems);
    edge_logits_max<<<(Etot + T - 1) / T, T, 0, stream>>>(src, dst, E, N, als, ald,
                                                          eb, mb);
    edge_softmax_sum<<<(Etot + T - 1) / T, T, 0, stream>>>(src, dst, E, N, eb, mb, sb);
    edge_aggregate<<<(int)((aggElems + T - 1) / T), T, 0, stream>>>(src, dst, E, N, hb,
                                                                    eb, sb, obuf);
    bias_gelu<<<(int)((nodeElems + T - 1) / T), T, 0, stream>>>(obuf, bl[layer], xnext,
                                                                nodeElems);
    xin = xnext;
  }
}